// CriterionIFV_85796266705552
// MI455X (gfx1250) — compile-verified
//
#include <hip/hip_runtime.h>
#include <hip/hip_bf16.h>

// ---------------------------------------------------------------------------
// CriterionIFV loss on MI455X (gfx1250).
// B=8, C=512, H=W=128, NUM_CLASSES=19 (padded to 32), IGNORE=255.
// Pass 1: class-sum histograms (float4 streaming + LDS ds_add_f32).
// Pass 2: per-pixel dot(feat, class-mean) for all 32 padded classes via
//         V_WMMA_F32_16X16X4_F32 (two 16-pixel A tiles share each B tile),
//         plus ||feat||^2, then cosine + masked-mean loss.
// ---------------------------------------------------------------------------

#define B_   8
#define C_   512
#define HW_  16384
#define NC_  19
#define KP_  32          // classes padded to 32 (two 16-wide WMMA N tiles)
#define IGN_ 255

typedef float v2f __attribute__((ext_vector_type(2)));
typedef float v8f __attribute__((ext_vector_type(8)));

// ---- workspace layout (floats) --------------------------------------------
#define WS_SUMS_S 0          // B*C*KP = 131072 (becomes meanS after k2a)
#define WS_SUMS_T 131072     // 131072          (becomes meanT)
#define WS_COUNTS 262144     // B*KP = 256
#define WS_MSQ_S  262400     // 256  ||mean||^2 per (b,k)
#define WS_MSQ_T  262656     // 256
#define WS_ACCUM  262912     // {loss_sum, mask_cnt}
#define WS_TOTAL  262914

// ===========================================================================
// Kernel 1: per-(b,c,k) sums of S and T + per-(b,k) counts.
// Grid: B * (C/4) * (HW/4096) = 4096 blocks, 256 threads. float4 streaming.
// ===========================================================================
__global__ void k1_class_sums(const float* __restrict__ S,
                              const float* __restrict__ T,
                              const int*   __restrict__ lab,
                              float* __restrict__ sumsS,
                              float* __restrict__ sumsT,
                              float* __restrict__ counts)
{
    __shared__ float binsS[4][KP_];
    __shared__ float binsT[4][KP_];
    __shared__ float binsC[KP_];

    const int tid  = threadIdx.x;
    const int bid  = blockIdx.x;
    const int b    = bid >> 9;          // 128 cgroups * 4 tiles per batch
    const int rem  = bid & 511;
    const int cg   = rem >> 2;          // channel group of 4
    const int tile = rem & 3;           // 4096-pixel tile

    if (tid < 128)      ((float*)binsS)[tid]       = 0.0f;
    else                ((float*)binsT)[tid - 128] = 0.0f;
    if (tid < KP_)      binsC[tid] = 0.0f;
    __syncthreads();

    const int lbase  = b * HW_ + tile * 4096;
    const bool doCnt = (cg == 0);

    #pragma unroll
    for (int cl = 0; cl < 4; ++cl) {
        const int c    = cg * 4 + cl;
        const int base = (b * C_ + c) * HW_ + tile * 4096;
        for (int i = tid * 4; i < 4096; i += 1024) {
            const int4   k4 = *(const int4*)  (lab + lbase + i);
            const float4 s4 = *(const float4*)(S   + base  + i);
            const float4 t4 = *(const float4*)(T   + base  + i);
            const int   kk[4] = { k4.x, k4.y, k4.z, k4.w };
            const float ss[4] = { s4.x, s4.y, s4.z, s4.w };
            const float tt[4] = { t4.x, t4.y, t4.z, t4.w };
            #pragma unroll
            for (int j = 0; j < 4; ++j) {
                const int k = kk[j];
                if ((unsigned)k < (unsigned)NC_) {
                    atomicAdd(&binsS[cl][k], ss[j]);
                    atomicAdd(&binsT[cl][k], tt[j]);
                    if (doCnt && cl == 0) atomicAdd(&binsC[k], 1.0f);
                }
            }
        }
    }
    __syncthreads();

    if (tid < 128) {
        const int cl = tid >> 5, kk = tid & 31;
        if (kk < NC_)
            atomicAdd(&sumsS[(b * C_ + cg * 4 + cl) * KP_ + kk], binsS[cl][kk]);
    } else {
        const int t2 = tid - 128, cl = t2 >> 5, kk = t2 & 31;
        if (kk < NC_)
            atomicAdd(&sumsT[(b * C_ + cg * 4 + cl) * KP_ + kk], binsT[cl][kk]);
    }
    if (doCnt && tid < NC_) atomicAdd(&counts[b * KP_ + tid], binsC[tid]);
}

// ===========================================================================
// Kernel 2a: normalize sums -> means in place. 512 blocks x 256 threads.
// ===========================================================================
__global__ void k2a_means(float* __restrict__ sumsS,
                          float* __restrict__ sumsT,
                          const float* __restrict__ counts)
{
    const int idx = blockIdx.x * 256 + threadIdx.x;   // 0 .. 131071
    const int b  = idx >> 14;                         // C_*KP_ = 16384
    const int kk = idx & 31;
    const float inv = 1.0f / (counts[b * KP_ + kk] + 1e-6f);
    sumsS[idx] *= inv;
    sumsT[idx] *= inv;
}

// ===========================================================================
// Kernel 2b: ||mean||^2 per (b,k) for S and T. 2 blocks x 256 threads.
// ===========================================================================
__global__ void k2b_mean_norms(const float* __restrict__ meanS,
                               const float* __restrict__ meanT,
                               float* __restrict__ msqS,
                               float* __restrict__ msqT)
{
    const int t  = threadIdx.x;            // 0..255 -> (b,kk)
    const int b  = t >> 5;
    const int kk = t & 31;
    const float* m = (blockIdx.x == 0) ? meanS : meanT;
    float acc = 0.0f;
    for (int c = 0; c < C_; ++c) {
        const float v = m[(b * C_ + c) * KP_ + kk];
        acc += v * v;
    }
    if (blockIdx.x == 0) msqS[b * KP_ + kk] = acc;
    else                 msqT[b * KP_ + kk] = acc;
}

// ===========================================================================
// Kernel 3: WMMA dot pass + cosine + loss accumulation.
// 1024 blocks x 128 threads (4 waves); each wave owns 32 pixels (two 16-px
// A tiles sharing every B tile).  A (16px x 4ch f32) x B (4ch x 16cls f32)
// -> D (16x16 f32); 2 N-tiles x 2 pixel tiles x {S,T} = 8 WMMA / iter.
// ===========================================================================
__global__ void k3_wmma_loss(const float* __restrict__ S,
                             const float* __restrict__ T,
                             const int*   __restrict__ lab,
                             const float* __restrict__ meanS,
                             const float* __restrict__ meanT,
                             const float* __restrict__ msqS,
                             const float* __restrict__ msqT,
                             float* __restrict__ accum)
{
    __shared__ float ls[4][2][32][KP_];   // [wave][S/T][pixel][class]  32KB
    __shared__ float lnf[4][2][2][32];    // [wave][S/T][half][pixel]    2KB
    __shared__ float blkAcc[2];

    const int tid  = threadIdx.x;
    const int w    = tid >> 5;
    const int lane = tid & 31;
    const int m    = lane & 15;           // pixel-in-tile / class column
    const int hi   = lane >> 4;           // half-wave -> channel offset 2*hi

    if (tid == 0) { blkAcc[0] = 0.0f; blkAcc[1] = 0.0f; }

    const int g     = blockIdx.x * 4 + w;     // 0..4095 groups of 32 pixels
    const int b     = g >> 9;                 // 512 groups per batch
    const int pbase = (g & 511) * 32;

    v8f dS00 = {}, dS01 = {}, dS10 = {}, dS11 = {};   // [pixTile][clsTile]
    v8f dT00 = {}, dT01 = {}, dT10 = {}, dT11 = {};
    float nfS0 = 0.0f, nfS1 = 0.0f, nfT0 = 0.0f, nfT1 = 0.0f;

    // feature base: channel (c + 2*hi), pixel (pbase + m)  [+16 for tile 1]
    const int fb = (b * C_ + 2 * hi) * HW_ + pbase + m;
    // mean-table base: row (c + 2*hi), padded-class column m
    const int mb = (b * C_ + 2 * hi) * KP_ + m;

    for (int c = 0; c < C_; c += 4) {
        const int ia = fb + c * HW_;
        const float aS00 = S[ia],      aS01 = S[ia + HW_];        // pix tile 0
        const float aS10 = S[ia + 16], aS11 = S[ia + 16 + HW_];   // pix tile 1
        const float aT00 = T[ia],      aT01 = T[ia + HW_];
        const float aT10 = T[ia + 16], aT11 = T[ia + 16 + HW_];
        if (c + 8 < C_) {                       // hide the 64KB channel stride
            __builtin_prefetch(&S[ia + 8 * HW_], 0, 1);
            __builtin_prefetch(&S[ia + 16 + 8 * HW_], 0, 1);
            __builtin_prefetch(&T[ia + 8 * HW_], 0, 1);
            __builtin_prefetch(&T[ia + 16 + 8 * HW_], 0, 1);
        }
        v2f aS0 = { aS00, aS01 };
        v2f aS1 = { aS10, aS11 };
        v2f aT0 = { aT00, aT01 };
        v2f aT1 = { aT10, aT11 };
        nfS0 += aS00 * aS00 + aS01 * aS01;
        nfS1 += aS10 * aS10 + aS11 * aS11;
        nfT0 += aT00 * aT00 + aT01 * aT01;
        nfT1 += aT10 * aT10 + aT11 * aT11;

        const int ib = mb + c * KP_;
        v2f bS0 = { meanS[ib],      meanS[ib + KP_] };        // classes 0..15
        v2f bS1 = { meanS[ib + 16], meanS[ib + KP_ + 16] };   // classes 16..31
        v2f bT0 = { meanT[ib],      meanT[ib + KP_] };
        v2f bT1 = { meanT[ib + 16], meanT[ib + KP_ + 16] };

        dS00 = __builtin_amdgcn_wmma_f32_16x16x4_f32(false, aS0, false, bS0,
                                                     (short)0, dS00, false, false);
        dS01 = __builtin_amdgcn_wmma_f32_16x16x4_f32(false, aS0, false, bS1,
                                                     (short)0, dS01, false, false);
        dS10 = __builtin_amdgcn_wmma_f32_16x16x4_f32(false, aS1, false, bS0,
                                                     (short)0, dS10, false, false);
        dS11 = __builtin_amdgcn_wmma_f32_16x16x4_f32(false, aS1, false, bS1,
                                                     (short)0, dS11, false, false);
        dT00 = __builtin_amdgcn_wmma_f32_16x16x4_f32(false, aT0, false, bT0,
                                                     (short)0, dT00, false, false);
        dT01 = __builtin_amdgcn_wmma_f32_16x16x4_f32(false, aT0, false, bT1,
                                                     (short)0, dT01, false, false);
        dT10 = __builtin_amdgcn_wmma_f32_16x16x4_f32(false, aT1, false, bT0,
                                                     (short)0, dT10, false, false);
        dT11 = __builtin_amdgcn_wmma_f32_16x16x4_f32(false, aT1, false, bT1,
                                                     (short)0, dT11, false, false);
    }

    // spill D tiles: D row M = r + 8*hi lives in VGPR r, col = lane%16.
    // pixel tile 0 -> rows 0..15, pixel tile 1 -> rows 16..31.
    #pragma unroll
    for (int r = 0; r < 8; ++r) {
        const int row = r + 8 * hi;
        ls[w][0][row][m]           = dS00[r];
        ls[w][0][row][m + 16]      = dS01[r];
        ls[w][0][row + 16][m]      = dS10[r];
        ls[w][0][row + 16][m + 16] = dS11[r];
        ls[w][1][row][m]           = dT00[r];
        ls[w][1][row][m + 16]      = dT01[r];
        ls[w][1][row + 16][m]      = dT10[r];
        ls[w][1][row + 16][m + 16] = dT11[r];
    }
    lnf[w][0][hi][m]      = nfS0;
    lnf[w][0][hi][m + 16] = nfS1;
    lnf[w][1][hi][m]      = nfT0;
    lnf[w][1][hi][m + 16] = nfT1;
    __syncthreads();

    {   // every lane gathers one pixel (32 pixels per wave)
        const int pix = lane;
        const int p   = pbase + pix;
        const int k   = lab[b * HW_ + p];
        const bool valid = ((unsigned)k < (unsigned)NC_);

        const float nS = lnf[w][0][0][pix] + lnf[w][0][1][pix];
        const float nT = lnf[w][1][0][pix] + lnf[w][1][1][pix];

        const float dotS = valid ? ls[w][0][pix][k] : nS;
        const float dotT = valid ? ls[w][1][pix][k] : nT;
        const float mS   = valid ? msqS[b * KP_ + k] : nS;
        const float mT   = valid ? msqT[b * KP_ + k] : nT;

        const float xnS = fmaxf(sqrtf(nS), 1e-8f);
        const float ynS = fmaxf(sqrtf(mS), 1e-8f);
        const float xnT = fmaxf(sqrtf(nT), 1e-8f);
        const float ynT = fmaxf(sqrtf(mT), 1e-8f);

        const float pcS = dotS / (xnS * ynS);
        const float pcT = dotT / (xnT * ynT);
        const float d   = pcS - pcT;
        const float msk = (k != IGN_) ? 1.0f : 0.0f;

        atomicAdd(&blkAcc[0], d * d * msk);
        atomicAdd(&blkAcc[1], msk);
    }
    __syncthreads();
    if (tid == 0) {
        atomicAdd(&accum[0], blkAcc[0]);
        atomicAdd(&accum[1], blkAcc[1]);
    }
}

// ===========================================================================
// Kernel 4: finalize scalar loss.
// ===========================================================================
__global__ void k4_finalize(const float* __restrict__ accum,
                            float* __restrict__ out)
{
    const float s = accum[0], cnt = accum[1];
    out[0] = (cnt > 0.0f) ? (s / fmaxf(cnt, 1.0f)) : 0.0f;
}

// ===========================================================================
extern "C" void kernel_launch(void* const* d_in, const int* in_sizes, int n_in,
                              void* d_out, int out_size, void* d_ws, size_t ws_size,
                              hipStream_t stream)
{
    const float* S   = (const float*)d_in[0];
    const float* T   = (const float*)d_in[1];
    const int*   lab = (const int*)d_in[2];
    float*       out = (float*)d_out;

    float* ws     = (float*)d_ws;
    float* sumsS  = ws + WS_SUMS_S;
    float* sumsT  = ws + WS_SUMS_T;
    float* counts = ws + WS_COUNTS;
    float* msqS   = ws + WS_MSQ_S;
    float* msqT   = ws + WS_MSQ_T;
    float* accum  = ws + WS_ACCUM;

    hipMemsetAsync(d_ws, 0, (size_t)WS_TOTAL * sizeof(float), stream);

    k1_class_sums <<<4096, 256, 0, stream>>>(S, T, lab, sumsS, sumsT, counts);
    k2a_means     <<<512,  256, 0, stream>>>(sumsS, sumsT, counts);
    k2b_mean_norms<<<2,    256, 0, stream>>>(sumsS, sumsT, msqS, msqT);
    k3_wmma_loss  <<<1024, 128, 0, stream>>>(S, T, lab, sumsS, sumsT,
                                             msqS, msqT, accum);
    k4_finalize   <<<1,    1,   0, stream>>>(accum, out);
}